// SparseKVCache_88562225643884
// MI455X (gfx1250) — compile-verified
//
#include <hip/hip_runtime.h>
#include <hip/hip_bf16.h>

// ---------------------------------------------------------------------------
// SparseKVCache top-k magnitude mask for gfx1250 (MI455X).
// key = float bits & 0x7FFFFFFF is monotone in |x|. 3-level radix select
// (bits [30:20], [19:9], [8:0]) finds exact 31-bit threshold T and tie
// remainder r. Mask pass keeps key>T plus first r ties (atomic counter),
// streamed through a per-wave async-to-LDS ring (CDNA5 ASYNCcnt path) with
// non-temporal output stores (input stays L2-resident; output is write-once).
// ---------------------------------------------------------------------------

#define WS_HIST 0
#define WS_SEL  2048
#define WS_KREM 2049
#define WS_TIE  2050
#define WS_WORDS 2051

#define MASK_WAVES_PER_BLOCK 8
#define MASK_NSTAGE 8

typedef float vfloat4 __attribute__((ext_vector_type(4)));

// ----- CDNA5 async helpers -------------------------------------------------

// LDS byte offset of a generic pointer to __shared__ (low 32 bits of the
// generic address are the group-segment offset on AMDGPU).
__device__ __forceinline__ unsigned lds_offset_of(const void* p) {
  return (unsigned)(unsigned long long)p;
}

// global_load_async_to_lds_b128 vdst(lds byte offset), vaddr(64b global), off
__device__ __forceinline__ void async_load_b128_to_lds(unsigned lds_off,
                                                       unsigned long long gaddr) {
  asm volatile("global_load_async_to_lds_b128 %0, %1, off"
               :: "v"(lds_off), "v"(gaddr) : "memory");
}

template <int N>
__device__ __forceinline__ void wait_asynccnt() {
#if __has_builtin(__builtin_amdgcn_s_wait_asynccnt)
  __builtin_amdgcn_s_wait_asynccnt(N);
#else
  asm volatile("s_wait_asynccnt %0" :: "i"(N) : "memory");
#endif
}

__device__ __forceinline__ void wait_dscnt0() {
  asm volatile("s_wait_dscnt 0" ::: "memory");
}

// ----- init: zero hist + state, seed krem = k ------------------------------

__global__ void skv_init(unsigned* __restrict__ ws, unsigned k) {
  int i = blockIdx.x * blockDim.x + threadIdx.x;
  int stride = blockDim.x * gridDim.x;
  for (; i < WS_WORDS; i += stride) ws[i] = (i == WS_KREM) ? k : 0u;
}

// ----- histogram pass (one radix level) ------------------------------------
// Counts elements whose (key >> cmp_shift) matches the selected prefix into
// buckets (key >> bshift) & bmask. LDS-privatized, flushed with global atomics.
// Default (RT) load policy: keeps the 128MiB tensor resident in the 192MB L2
// so levels 1/2 and the mask pass read from L2, not HBM.

__global__ void __launch_bounds__(256)
skv_hist(const float4* __restrict__ x4, long n4,
         const float* __restrict__ x, long total,
         unsigned* __restrict__ ws,
         int cmp_shift, int bshift, unsigned bmask) {
  __shared__ unsigned lhist[2048];
  for (int i = threadIdx.x; i < 2048; i += blockDim.x) lhist[i] = 0u;
  __syncthreads();

  const unsigned sel = ws[WS_SEL];
  const unsigned cmp = sel >> cmp_shift;   // level 0: sel==0, cmp_shift=31 -> 0

  long tid    = (long)blockIdx.x * blockDim.x + threadIdx.x;
  long stride = (long)gridDim.x * blockDim.x;
  for (long i = tid; i < n4; i += stride) {
    float4 v = x4[i];
    unsigned kk[4] = { __float_as_uint(v.x), __float_as_uint(v.y),
                       __float_as_uint(v.z), __float_as_uint(v.w) };
#pragma unroll
    for (int c = 0; c < 4; ++c) {
      unsigned key = kk[c] & 0x7fffffffu;
      if ((key >> cmp_shift) == cmp)
        atomicAdd(&lhist[(key >> bshift) & bmask], 1u);
    }
  }
  // ragged tail (total % 4), handled by block 0 only
  if (blockIdx.x == 0) {
    for (long i = n4 * 4 + threadIdx.x; i < total; i += blockDim.x) {
      unsigned key = __float_as_uint(x[i]) & 0x7fffffffu;
      if ((key >> cmp_shift) == cmp)
        atomicAdd(&lhist[(key >> bshift) & bmask], 1u);
    }
  }
  __syncthreads();
  for (int i = threadIdx.x; i < 2048; i += blockDim.x)
    if (lhist[i]) atomicAdd(&ws[WS_HIST + i], lhist[i]);
}

// ----- scan: pick bucket from the top, update prefix/krem, clear hist ------

__global__ void skv_scan(unsigned* __restrict__ ws, int nbuckets, int shift) {
  if (blockIdx.x == 0 && threadIdx.x == 0) {
    unsigned krem = ws[WS_KREM];
    unsigned cum = 0;
    int b = 0;
    for (int i = nbuckets - 1; i >= 0; --i) {
      unsigned c = ws[WS_HIST + i];
      if (cum + c >= krem) { b = i; break; }
      cum += c;
    }
    ws[WS_SEL] |= ((unsigned)b) << shift;
    ws[WS_KREM] = krem - cum;            // ties to keep within bucket b, >= 1
    for (int i = 0; i < nbuckets; ++i) ws[WS_HIST + i] = 0u;
  }
}

// ----- mask pass: async-to-LDS staged stream, NT output stores -------------

__device__ __forceinline__ float keep_val(float v, unsigned T, unsigned r,
                                          unsigned* tie) {
  unsigned key = __float_as_uint(v) & 0x7fffffffu;
  if (key > T) return v;
  if (key == T) { if (atomicAdd(tie, 1u) < r) return v; }
  return 0.0f;
}

__global__ void __launch_bounds__(256)
skv_mask(const float4* __restrict__ x4, long n4,
         const float* __restrict__ x, long total,
         vfloat4* __restrict__ o4, float* __restrict__ o,
         unsigned* __restrict__ ws) {
  __shared__ float4 stage[MASK_WAVES_PER_BLOCK][MASK_NSTAGE][32];

  const unsigned T = ws[WS_SEL];
  const unsigned r = ws[WS_KREM];
  unsigned* tie = &ws[WS_TIE];

  const int lane = threadIdx.x & 31;
  const int w    = threadIdx.x >> 5;
  const long gwave  = (long)blockIdx.x * MASK_WAVES_PER_BLOCK + w;
  const long nwaves = (long)gridDim.x * MASK_WAVES_PER_BLOCK;
  const long ntiles = (n4 + 31) / 32;     // one tile = 32 lanes x float4

  // ragged tail (total % 4), scalar, one wave
  if (blockIdx.x == 0 && w == 0) {
    for (long i = n4 * 4 + lane; i < total; i += 32) {
      float ov = keep_val(x[i], T, r, tie);
      __builtin_nontemporal_store(ov, &o[i]);
    }
  }

  long issueT = gwave, consumeT = gwave;
  // prime the ring
  for (int s = 0; s < MASK_NSTAGE && issueT < ntiles; ++s) {
    long f = issueT * 32 + lane;
    if (f < n4)
      async_load_b128_to_lds(lds_offset_of(&stage[w][s][lane]),
                             (unsigned long long)(const void*)(x4 + f));
    issueT += nwaves;
  }
  int cs = 0;
  // steady state: 8 tiles in flight per wave on ASYNCcnt
  while (issueT < ntiles) {
    wait_asynccnt<MASK_NSTAGE - 1>();            // oldest tile landed in LDS
    long f = consumeT * 32 + lane;
    if (f < n4) {
      float4 v = stage[w][cs][lane];
      vfloat4 ov;
      ov.x = keep_val(v.x, T, r, tie);
      ov.y = keep_val(v.y, T, r, tie);
      ov.z = keep_val(v.z, T, r, tie);
      ov.w = keep_val(v.w, T, r, tie);
      __builtin_nontemporal_store(ov, o4 + f);
    }
    wait_dscnt0();   // DS reads of slot cs done before async engine reuses it
    long fi = issueT * 32 + lane;
    if (fi < n4)
      async_load_b128_to_lds(lds_offset_of(&stage[w][cs][lane]),
                             (unsigned long long)(const void*)(x4 + fi));
    consumeT += nwaves; issueT += nwaves;
    cs = (cs + 1) & (MASK_NSTAGE - 1);
  }
  // drain
  wait_asynccnt<0>();
  while (consumeT < ntiles) {
    long f = consumeT * 32 + lane;
    if (f < n4) {
      float4 v = stage[w][cs][lane];
      vfloat4 ov;
      ov.x = keep_val(v.x, T, r, tie);
      ov.y = keep_val(v.y, T, r, tie);
      ov.z = keep_val(v.z, T, r, tie);
      ov.w = keep_val(v.w, T, r, tie);
      __builtin_nontemporal_store(ov, o4 + f);
    }
    consumeT += nwaves;
    cs = (cs + 1) & (MASK_NSTAGE - 1);
  }
}

// ---------------------------------------------------------------------------

extern "C" void kernel_launch(void* const* d_in, const int* in_sizes, int n_in,
                              void* d_out, int out_size, void* d_ws, size_t ws_size,
                              hipStream_t stream) {
  const float* x = (const float*)d_in[0];
  long total = (long)in_sizes[0];
  long n4 = total / 4;
  // elements to keep = int(total * (1 - 0.5))
  unsigned k = (unsigned)((double)total * 0.5);
  unsigned* ws = (unsigned*)d_ws;
  float* out = (float*)d_out;

  skv_init<<<4, 256, 0, stream>>>(ws, k);
  // level 0: bits [30:20]  (HBM read; warms L2)
  skv_hist<<<2048, 256, 0, stream>>>((const float4*)x, n4, x, total, ws, 31, 20, 0x7FFu);
  skv_scan<<<1, 1, 0, stream>>>(ws, 2048, 20);
  // level 1: bits [19:9]   (L2-resident read)
  skv_hist<<<2048, 256, 0, stream>>>((const float4*)x, n4, x, total, ws, 20, 9, 0x7FFu);
  skv_scan<<<1, 1, 0, stream>>>(ws, 2048, 9);
  // level 2: bits [8:0]    (L2-resident read)
  skv_hist<<<2048, 256, 0, stream>>>((const float4*)x, n4, x, total, ws, 9, 0, 0x1FFu);
  skv_scan<<<1, 1, 0, stream>>>(ws, 512, 0);
  // final mask: async-to-LDS staged stream, NT stores to HBM
  skv_mask<<<2048, 256, 0, stream>>>((const float4*)x, n4, x, total,
                                     (vfloat4*)out, out, ws);
}